// TopKSiLU_19464791786012
// MI455X (gfx1250) — compile-verified
//
#include <hip/hip_runtime.h>
#include <hip/hip_bf16.h>

#define TDIM  4096
#define KSEL  2048
#define TPB   256
#define F4PT  4          // float4 chunks per thread: TDIM / 4 / TPB

typedef float v4f __attribute__((ext_vector_type(4)));

__global__ __launch_bounds__(TPB)
void topk_silu_radix_kernel(const float* __restrict__ x, float* __restrict__ out)
{
    __shared__ __align__(16) float lrow[TDIM];   // 16 KB row staging (async target)
    __shared__ unsigned hist[256];               // radix histogram
    __shared__ unsigned ssel[2];                 // {selected byte, new k}

    const int tid = threadIdx.x;
    const long long row = blockIdx.x;
    const float* __restrict__ xrow = x + row * (long long)TDIM;
    float* __restrict__ orow = out + row * (long long)TDIM;

    // LDS byte offset of lrow: low 32 bits of the generic (flat) pointer hold
    // the LDS-relative address in the shared aperture (ISA 10.2).
    unsigned lrow_base = (unsigned)(unsigned long long)&lrow[0];
    unsigned long long gbase = (unsigned long long)xrow;

    // ---- async copy of this row into LDS (gfx1250 async engine, ASYNCcnt),
    //      non-temporal: row is consumed once, keep it out of hot L2 ----
#pragma unroll
    for (int c = 0; c < F4PT; ++c) {
        unsigned boff = (unsigned)(tid + c * TPB) * 16u;   // byte offset in row
        asm volatile("global_load_async_to_lds_b128 %0, %1, %2 th:TH_LOAD_NT"
                     :
                     : "v"(lrow_base + boff), "v"(boff), "s"(gbase)
                     : "memory");
    }
    asm volatile("s_wait_asynccnt 0" ::: "memory");
    __syncthreads();

    // ---- compute h = silu(x) and |h| bit patterns, kept in registers ----
    float    hval[4 * F4PT];
    unsigned abits[4 * F4PT];
#pragma unroll
    for (int c = 0; c < F4PT; ++c) {
        float4 v = reinterpret_cast<const float4*>(lrow)[tid + c * TPB];
        float h0 = v.x / (1.0f + expf(-v.x));
        float h1 = v.y / (1.0f + expf(-v.y));
        float h2 = v.z / (1.0f + expf(-v.z));
        float h3 = v.w / (1.0f + expf(-v.w));
        hval[4*c+0] = h0; abits[4*c+0] = __float_as_uint(h0) & 0x7fffffffu;
        hval[4*c+1] = h1; abits[4*c+1] = __float_as_uint(h1) & 0x7fffffffu;
        hval[4*c+2] = h2; abits[4*c+2] = __float_as_uint(h2) & 0x7fffffffu;
        hval[4*c+3] = h3; abits[4*c+3] = __float_as_uint(h3) & 0x7fffffffu;
    }

    // ---- 4-pass radix select: exact K-th largest |h| (bit pattern).
    //      Scan+select done by a single wave32 (lane owns 8 bins): 3 barriers/pass.
    unsigned prefix = 0u, himask = 0u, kcur = KSEL;
#pragma unroll
    for (int pass = 0; pass < 4; ++pass) {
        const int shift = 24 - 8 * pass;

        hist[tid] = 0u;
        __syncthreads();

#pragma unroll
        for (int i = 0; i < 4 * F4PT; ++i) {
            unsigned b = abits[i];
            if ((b & himask) == prefix)
                atomicAdd(&hist[(b >> shift) & 0xFFu], 1u);
        }
        __syncthreads();

        if (tid < 32) {                      // wave 0 only: suffix scan + select
            unsigned b8[8];
            unsigned sumL = 0u;
#pragma unroll
            for (int j = 0; j < 8; ++j) { b8[j] = hist[tid * 8 + j]; sumL += b8[j]; }

            // wave suffix-inclusive scan over lane sums (higher lane = higher bins)
            unsigned v = sumL;
#pragma unroll
            for (int s = 1; s < 32; s <<= 1) {
                unsigned t = __shfl_down(v, s, 32);
                if (tid + s < 32) v += t;
            }
            unsigned above = v - sumL;       // count with byte >= (lane+1)*8
            if (above < kcur && v >= kcur) { // exactly one lane holds rank kcur
                unsigned cum = above;
#pragma unroll
                for (int j = 7; j >= 0; --j) {
                    unsigned ncum = cum + b8[j];
                    if (ncum >= kcur) {
                        ssel[0] = (unsigned)(tid * 8 + j);  // selected byte
                        ssel[1] = kcur - cum;               // new rank inside bin
                        break;
                    }
                    cum = ncum;
                }
            }
        }
        __syncthreads();

        prefix |= ssel[0] << shift;
        kcur    = ssel[1];
        himask |= 0xFFu << shift;
    }
    const unsigned thr = prefix;   // bits of the K-th largest |h|

    // ---- apply threshold, streaming non-temporal float4 stores ----
#pragma unroll
    for (int c = 0; c < F4PT; ++c) {
        v4f o4;
        o4.x = (abits[4*c+0] >= thr) ? hval[4*c+0] : 0.0f;
        o4.y = (abits[4*c+1] >= thr) ? hval[4*c+1] : 0.0f;
        o4.z = (abits[4*c+2] >= thr) ? hval[4*c+2] : 0.0f;
        o4.w = (abits[4*c+3] >= thr) ? hval[4*c+3] : 0.0f;
        __builtin_nontemporal_store(o4, reinterpret_cast<v4f*>(orow) + tid + c * TPB);
    }
}

extern "C" void kernel_launch(void* const* d_in, const int* in_sizes, int n_in,
                              void* d_out, int out_size, void* d_ws, size_t ws_size,
                              hipStream_t stream)
{
    const float* x = (const float*)d_in[0];
    float* out = (float*)d_out;
    const int rows = in_sizes[0] / TDIM;   // 4 * 4096 = 16384
    topk_silu_radix_kernel<<<rows, TPB, 0, stream>>>(x, out);
}